// Encoder_5781025980487
// MI455X (gfx1250) — compile-verified
//
#include <hip/hip_runtime.h>
#include <hip/hip_bf16.h>

#define IN_CH   512
#define OUT_CH  64
#define KCHUNK  128   // K-tile staged in LDS: 128*64*4B = 32 KB
#define ROWS_PER_BLOCK 128  // 8 waves * 16 rows

typedef __attribute__((ext_vector_type(2))) float v2f;
typedef __attribute__((ext_vector_type(4))) float v4f;
typedef __attribute__((ext_vector_type(8))) float v8f;

// ---------------------------------------------------------------------------
// k0: zero the aggregation buffer, init degree to 1.0 (self loop)
// ---------------------------------------------------------------------------
__global__ void init_kernel(float* __restrict__ agg, float* __restrict__ deg, int n) {
    int i = blockIdx.x * blockDim.x + threadIdx.x;
    int total = n * OUT_CH;
    if (i < total) agg[i] = 0.0f;
    if (i < n)     deg[i] = 1.0f;   // self loop contributes 1 to every dst degree
}

// ---------------------------------------------------------------------------
// k1: degree over edge destinations
// ---------------------------------------------------------------------------
__global__ void deg_kernel(const int* __restrict__ dst, float* __restrict__ deg, int e) {
    int i = blockIdx.x * blockDim.x + threadIdx.x;
    if (i < e) atomicAdd(&deg[dst[i]], 1.0f);
}

// ---------------------------------------------------------------------------
// k2: dinv = rsqrt(deg)  (deg >= 1 always, so no zero guard needed)
// ---------------------------------------------------------------------------
__global__ void dinv_kernel(float* __restrict__ deg, int n) {
    int i = blockIdx.x * blockDim.x + threadIdx.x;
    if (i < n) deg[i] = rsqrtf(deg[i]);
}

// ---------------------------------------------------------------------------
// k3: h = x @ W  via V_WMMA_F32_16X16X4_F32 (full fp32; GEMM is BW-bound on
// the 205 MB x read, so fp32 matrix ops are free and exact).
// 256 threads = 8 waves; wave w computes rows [row0, row0+16) x all 64 cols
// as four 16x16 C tiles (32 accumulator VGPRs).
// W is staged in LDS *K-pair-packed*: element j = p*64+n holds the float2
// (W[2p][n], W[2p+1][n]), so every B-fragment is one aligned ds_load_b64
// directly into an even VGPR pair (no v_dual_mov shuffles before the WMMAs).
// ---------------------------------------------------------------------------
__global__ __launch_bounds__(256) void gemm_wmma_kernel(
        const float* __restrict__ x, const float* __restrict__ W,
        float* __restrict__ h, int n) {
    __shared__ __align__(16) v2f lds_w[(KCHUNK / 2) * OUT_CH];   // 32 KB

    const int tid   = threadIdx.x;
    const int wave  = tid >> 5;
    const int lane  = tid & 31;
    const int half  = lane >> 4;     // 0: lanes 0-15, 1: lanes 16-31
    const int l15   = lane & 15;
    const int kh    = half * 2;      // K-half offset for A/B fragments
    const int row0  = blockIdx.x * ROWS_PER_BLOCK + wave * 16;

    // A-fragment row for this lane (M = l15 in both half-waves); clamp so
    // tail-block loads stay in bounds without breaking EXEC=all-ones.
    int arow = row0 + l15;
    if (arow >= n) arow = n - 1;
    const float* xrow = x + (size_t)arow * IN_CH;

    v8f acc[4] = {{}, {}, {}, {}};

    for (int kc = 0; kc < IN_CH; kc += KCHUNK) {
        // Stage W[kc .. kc+127][0..63] K-pair-packed. 4096 float2 elements,
        // 256 threads -> 16 each. Consecutive tids hit consecutive n within a
        // row pair, so both global reads stay coalesced.
        #pragma unroll
        for (int i = 0; i < (KCHUNK / 2) * OUT_CH / 256; ++i) {
            int j = tid + i * 256;
            int p = j >> 6;           // K-pair index 0..63
            int c = j & 63;           // output channel
            v2f w2;
            w2.x = W[(size_t)(kc + 2 * p)     * OUT_CH + c];
            w2.y = W[(size_t)(kc + 2 * p + 1) * OUT_CH + c];
            lds_w[j] = w2;
        }
        __syncthreads();

        #pragma unroll 4
        for (int k = 0; k < KCHUNK; k += 4) {
            v2f a;
            a.x = xrow[kc + k + kh];
            a.y = xrow[kc + k + kh + 1];
            const int prow = (k >> 1) + half;   // K-pair row for this half-wave
            #pragma unroll
            for (int t = 0; t < 4; ++t) {
                v2f b = lds_w[prow * OUT_CH + t * 16 + l15];  // one ds_load_b64
                // 8 args: (neg_a, A, neg_b, B, c_mod, C, reuse_a, reuse_b)
                acc[t] = __builtin_amdgcn_wmma_f32_16x16x4_f32(
                    false, a, false, b, (short)0, acc[t], false, false);
            }
        }
        __syncthreads();
    }

    // C/D layout: VGPR v holds M = half*8 + v, N = l15 (per 16x16 tile)
    #pragma unroll
    for (int v = 0; v < 8; ++v) {
        int row = row0 + half * 8 + v;
        if (row < n) {
            #pragma unroll
            for (int t = 0; t < 4; ++t)
                h[(size_t)row * OUT_CH + t * 16 + l15] = acc[t][v];
        }
    }
}

// ---------------------------------------------------------------------------
// k4: edge scatter. One wave32 per edge, 2 channels per lane.
// h and agg are 25.6 MB each -> L2-resident; atomics resolve in L2.
// ---------------------------------------------------------------------------
__global__ __launch_bounds__(256) void scatter_kernel(
        const float* __restrict__ h, const int* __restrict__ src,
        const int* __restrict__ dst, const float* __restrict__ dinv,
        float* __restrict__ agg, int e) {
    int edge = blockIdx.x * 8 + (threadIdx.x >> 5);
    if (edge >= e) return;
    int lane = threadIdx.x & 31;
    int s = src[edge];
    int d = dst[edge];
    float norm = dinv[s] * dinv[d];
    const float2* hs = (const float2*)(h + (size_t)s * OUT_CH);
    float2 v = hs[lane];
    float* out = agg + (size_t)d * OUT_CH + lane * 2;
    atomicAdd(out,     v.x * norm);
    atomicAdd(out + 1, v.y * norm);
}

// ---------------------------------------------------------------------------
// k5: out = PReLU(agg + h*dinv^2 (self loop) + bias)
// ---------------------------------------------------------------------------
__global__ void finalize_kernel(const float* __restrict__ agg,
                                const float* __restrict__ h,
                                const float* __restrict__ dinv,
                                const float* __restrict__ bias,
                                const float* __restrict__ alpha,
                                float* __restrict__ out, int n) {
    int idx = blockIdx.x * blockDim.x + threadIdx.x;
    if (idx >= n * OUT_CH) return;
    int c = idx & (OUT_CH - 1);
    int i = idx >> 6;
    float di = dinv[i];
    float v = agg[idx] + h[idx] * di * di + bias[c];
    out[idx] = (v >= 0.0f) ? v : alpha[c] * v;
}

// ---------------------------------------------------------------------------
extern "C" void kernel_launch(void* const* d_in, const int* in_sizes, int n_in,
                              void* d_out, int out_size, void* d_ws, size_t ws_size,
                              hipStream_t stream) {
    const float* x     = (const float*)d_in[0];
    const int*   ei    = (const int*)  d_in[1];   // [2, E] row-major
    const float* W     = (const float*)d_in[2];
    const float* bias  = (const float*)d_in[3];
    const float* alpha = (const float*)d_in[4];
    float*       out   = (float*)d_out;

    const int N = in_sizes[0] / IN_CH;
    const int E = in_sizes[1] / 2;
    const int* src = ei;
    const int* dst = ei + E;

    // workspace layout (256B-aligned): h [N*64], agg [N*64], deg/dinv [N]
    char* ws = (char*)d_ws;
    size_t hw_bytes = ((size_t)N * OUT_CH * sizeof(float) + 255) & ~(size_t)255;
    float* h    = (float*)ws;
    float* agg  = (float*)(ws + hw_bytes);
    float* dinv = (float*)(ws + 2 * hw_bytes);   // holds deg, then dinv in place

    const int threads = 256;
    int totNC = N * OUT_CH;

    init_kernel<<<(totNC + threads - 1) / threads, threads, 0, stream>>>(agg, dinv, N);
    deg_kernel<<<(E + threads - 1) / threads, threads, 0, stream>>>(dst, dinv, E);
    dinv_kernel<<<(N + threads - 1) / threads, threads, 0, stream>>>(dinv, N);
    gemm_wmma_kernel<<<(N + ROWS_PER_BLOCK - 1) / ROWS_PER_BLOCK, threads, 0, stream>>>(
        x, W, h, N);
    scatter_kernel<<<(E + 7) / 8, threads, 0, stream>>>(h, src, dst, dinv, agg, E);
    finalize_kernel<<<(totNC + threads - 1) / threads, threads, 0, stream>>>(
        agg, h, dinv, bias, alpha, out, N);
}